// MaskedTriplaneSampler_84507776516822
// MI455X (gfx1250) — compile-verified
//
#include <hip/hip_runtime.h>
#include <stdint.h>

namespace {
constexpr int kB = 4, kR = 4096, kP = 64, kC = 32, kH = 256, kW = 256;
constexpr int kHW = kH * kW;                  // 65536 floats per channel plane
constexpr int kPoints = kB * kR * kP;         // 1,048,576
constexpr int kBlock = 256;
constexpr int kGrid = kPoints / kBlock;       // 4096 (exact)
}

typedef __attribute__((ext_vector_type(2))) float        f32x2;
typedef __attribute__((ext_vector_type(4))) float        f32x4;
typedef __attribute__((ext_vector_type(4))) unsigned int u32x4;
typedef __attribute__((ext_vector_type(8))) int          i32x8;
typedef __attribute__((ext_vector_type(4))) int          i32x4;

// Per-plane bilinear setup with pair-load (b64) addressing.
// The two x-corners are adjacent in memory, so each bilinear row is one
// global_load_b64. Pair base xb = clamp(x0, 0, W-2); when x0 != xb (x0 == -1
// or x0 == W-1) the valid corner lands in the other half of the pair, so the
// two x-weights are swapped once here (the invalid corner's weight is 0).
// Validity is folded into the weights; matches grid_sample(bilinear,
// align_corners=False, padding_mode='zeros').
__device__ __forceinline__ void plane_setup(float x, float y, uint32_t planeBase,
                                            float w[4], uint32_t o[2]) {
  float fx = (x + 1.0f) * 128.0f - 0.5f;   // (x+1)*W/2 - 0.5, W=256
  float fy = (y + 1.0f) * 128.0f - 0.5f;
  float x0f = floorf(fx), y0f = floorf(fy);
  float wx = fx - x0f,  wy = fy - y0f;
  int x0 = (int)x0f, y0 = (int)y0f;
  int x1 = x0 + 1,   y1 = y0 + 1;
  // x-direction weights with validity folded in
  float u0 = (1.0f - wx) * ((x0 >= 0 && x0 < kW) ? 1.0f : 0.0f);
  float u1 = wx          * ((x1 >= 0 && x1 < kW) ? 1.0f : 0.0f);
  // y-direction weights with validity folded in
  float s0 = (1.0f - wy) * ((y0 >= 0 && y0 < kH) ? 1.0f : 0.0f);
  float s1 = wy          * ((y1 >= 0 && y1 < kH) ? 1.0f : 0.0f);
  int xb  = min(max(x0, 0), kW - 2);     // pair base: loads x = xb, xb+1
  int yc0 = min(max(y0, 0), kH - 1);
  int yc1 = min(max(y1, 0), kH - 1);
  bool swp = (x0 != xb);                 // pair shifted -> swap x-weights
  float A  = swp ? u1 : u0;              // weight for pair.x
  float Bw = swp ? u0 : u1;              // weight for pair.y
  w[0] = A * s0;  w[1] = Bw * s0;        // row y0
  w[2] = A * s1;  w[3] = Bw * s1;        // row y1
  o[0] = planeBase + (uint32_t)(yc0 * kW + xb);
  o[1] = planeBase + (uint32_t)(yc1 * kW + xb);
}

__global__ __launch_bounds__(kBlock) void triplane_sample_kernel(
    const float* __restrict__ origins,     // [B,R,3]
    const float* __restrict__ directions,  // [B,R,3]
    const float* __restrict__ lengths,     // [B,R,P]
    const float* __restrict__ features,    // [B,3,C,H,W]
    const float* __restrict__ w2l,         // [B,4,4]
    float* __restrict__ out_feats,         // [B,R,P,C]
    float* __restrict__ out_mask)          // [B,R,P,1]
{
  // ---- CDNA5 TDM: async L2 pre-warm of an 8KB features chunk into LDS scratch.
  // Fire-and-forget (nothing reads s_tdm); retired with s_wait_tensorcnt at end.
  __shared__ __align__(16) char s_tdm[8192];
#if defined(__gfx1250__) && __has_builtin(__builtin_amdgcn_tensor_load_to_lds)
  if (threadIdx.x < 32) {   // one wave issues the TDM op (TDM ignores EXEC)
    const uint32_t chunk = 2048;  // floats (8KB)
    uint64_t ga = (uint64_t)(uintptr_t)(features + (size_t)blockIdx.x * chunk);
    uint32_t lds_off = (uint32_t)(uintptr_t)(&s_tdm[0]); // LDS offset = addr[31:0]
    u32x4 g0;
    g0[0] = 1u;                                  // count=1, user mode
    g0[1] = lds_off;                             // lds_addr (bytes)
    g0[2] = (uint32_t)ga;                        // global_addr[31:0]
    g0[3] = (uint32_t)((ga >> 32) & 0x1FFFFFFu)  // global_addr[56:32]
          | (2u << 30);                          // type = 2 ("image")
    i32x8 g1;
    g1[0] = 0x20000;                      // data_size=2 (4B), mask=0, no pad/iter
    g1[1] = (int)((chunk & 0xFFFFu) << 16);        // tensor_dim0 lo16
    g1[2] = (int)((chunk >> 16) | (1u << 16));     // tensor_dim0 hi16 | tensor_dim1=1
    g1[3] = (int)(chunk << 16);                    // tile_dim0 = 2048
    g1[4] = 1;                                     // tile_dim1=1, tile_dim2=0
    g1[5] = (int)chunk;                            // tensor_dim0_stride lo32
    g1[6] = (int)(chunk << 16);                    // ..hi16=0 | tensor_dim1_stride lo16
    g1[7] = 0;                                     // tensor_dim1_stride hi32
    i32x4 g2; g2[0] = 1; g2[1] = 1; g2[2] = 0; g2[3] = 0;   // tensor_dim2/3=1
    i32x4 g3; g3[0] = 0; g3[1] = (int)(1u << 16); g3[2] = 0; g3[3] = 0; // tensor_dim4=1
    i32x8 g4; g4[0]=0; g4[1]=0; g4[2]=0; g4[3]=0; g4[4]=0; g4[5]=0; g4[6]=0; g4[7]=0;
    // clang-23 / therock-10.0 form: 6 args, extra int32x8 zero-filled.
    __builtin_amdgcn_tensor_load_to_lds(g0, g1, g2, g3, g4, 0);
  }
#endif

  const int p   = blockIdx.x * kBlock + threadIdx.x;  // point id, exact cover
  const int ray = p >> 6;                             // P = 64
  const int b   = p >> 18;                            // R*P = 2^18

  // Ray data (rays shared by 64 threads; L0/L2 absorb the reuse).
  const float t  = lengths[p];
  const float ox = origins[ray * 3 + 0], oy = origins[ray * 3 + 1], oz = origins[ray * 3 + 2];
  const float dx = directions[ray * 3 + 0], dy = directions[ray * 3 + 1], dz = directions[ray * 3 + 2];
  const float* M = w2l + b * 16;   // uniform within a block

  // o_local[j] = sum_i o[i]*M[i][j] + M[3][j] ; d_local[j] = sum_i d[i]*M[i][j]
  const float ol0 = ox * M[0] + oy * M[4] + oz * M[8]  + M[12];
  const float ol1 = ox * M[1] + oy * M[5] + oz * M[9]  + M[13];
  const float ol2 = ox * M[2] + oy * M[6] + oz * M[10] + M[14];
  const float dl0 = dx * M[0] + dy * M[4] + dz * M[8];
  const float dl1 = dx * M[1] + dy * M[5] + dz * M[9];
  const float dl2 = dx * M[2] + dy * M[6] + dz * M[10];

  const float q0 = ol0 + dl0 * t;
  const float q1 = ol1 + dl1 * t;
  const float q2 = ol2 + dl2 * t;

  const float mask = (q0 >= -1.0f && q0 <= 1.0f &&
                      q1 >= -1.0f && q1 <= 1.0f &&
                      q2 >= -1.0f && q2 <= 1.0f) ? 1.0f : 0.0f;

  // Per-plane channel-0 bases: ((b*3 + k) * C) * H*W
  const uint32_t base0 = (uint32_t)(b * 3 + 0) * (uint32_t)(kC * kHW);
  const uint32_t base1 = (uint32_t)(b * 3 + 1) * (uint32_t)(kC * kHW);
  const uint32_t base2 = (uint32_t)(b * 3 + 2) * (uint32_t)(kC * kHW);

  float    w[3][4];
  uint32_t o[3][2];
  plane_setup(q0, q1, base0, w[0], o[0]);  // plane 0: dims (0,1)
  plane_setup(q0, q2, base1, w[1], o[1]);  // plane 1: dims (0,2)
  plane_setup(q1, q2, base2, w[2], o[2]);  // plane 2: dims (1,2)

  // Fully unrolled channel loop: 32 channels x 3 planes x 2 row-pair b64 loads.
  // Channel stride 65536 floats = 256KB -> fits the 24-bit VMEM inst offset,
  // so each row-pair keeps one VGPR offset across all 32 channel loads.
  float acc[kC];
#pragma unroll
  for (int c = 0; c < kC; ++c) {
    const uint32_t coff = (uint32_t)c * (uint32_t)kHW;
    float a = 0.0f;
#pragma unroll
    for (int k = 0; k < 3; ++k) {
      f32x2 r0 = *(const f32x2*)(features + (o[k][0] + coff));  // row y0: x, x+1
      f32x2 r1 = *(const f32x2*)(features + (o[k][1] + coff));  // row y1: x, x+1
      a = fmaf(w[k][0], r0.x, a);
      a = fmaf(w[k][1], r0.y, a);
      a = fmaf(w[k][2], r1.x, a);
      a = fmaf(w[k][3], r1.y, a);
    }
    acc[c] = a;
  }

  // Streaming output: non-temporal (TH=NT) so the 132MB result does not evict
  // the L2-resident 96MB features tensor. Each lane writes 128B contiguous;
  // the wave covers a contiguous 4KB span -> full-line write combining.
  f32x4* outv = (f32x4*)(out_feats + (size_t)p * kC);
#pragma unroll
  for (int i = 0; i < kC / 4; ++i) {
    f32x4 v = { acc[4 * i + 0], acc[4 * i + 1], acc[4 * i + 2], acc[4 * i + 3] };
    __builtin_nontemporal_store(v, outv + i);
  }
  __builtin_nontemporal_store(mask, out_mask + p);

#if defined(__gfx1250__) && __has_builtin(__builtin_amdgcn_s_wait_tensorcnt)
  __builtin_amdgcn_s_wait_tensorcnt(0);   // retire the TDM prefetch
#endif
}

extern "C" void kernel_launch(void* const* d_in, const int* in_sizes, int n_in,
                              void* d_out, int out_size, void* d_ws, size_t ws_size,
                              hipStream_t stream) {
  const float* origins    = (const float*)d_in[0];
  const float* directions = (const float*)d_in[1];
  const float* lengths    = (const float*)d_in[2];
  const float* features   = (const float*)d_in[3];
  const float* w2l        = (const float*)d_in[4];

  float* out_feats = (float*)d_out;                              // [B,R,P,C]
  float* out_mask  = (float*)d_out + (size_t)kPoints * kC;       // [B,R,P,1]

  hipLaunchKernelGGL(triplane_sample_kernel, dim3(kGrid), dim3(kBlock), 0, stream,
                     origins, directions, lengths, features, w2l,
                     out_feats, out_mask);
}